// LearnedRouter_1726576855450
// MI455X (gfx1250) — compile-verified
//
#include <hip/hip_runtime.h>
#include <hip/hip_bf16.h>

// Problem constants (match reference).
#define NTOK 16384
#define HDIM 4096
#define NEXP 64

typedef float v2f __attribute__((ext_vector_type(2)));
typedef float v8f __attribute__((ext_vector_type(8)));

static __device__ __forceinline__ v2f ldg_nt2(const float* p) {
    return __builtin_nontemporal_load((const v2f*)p);
}
static __device__ __forceinline__ v2f ldg2(const float* p) {
    return *(const v2f*)p;
}

// ---------------------------------------------------------------------------
// Kernel 1: logits[N,E] = x[N,H] @ W[E,H]^T with V_WMMA_F32_16X16X4_F32.
// One wave -> 16 rows x 64 experts (4 accumulator tiles of 16x16 f32);
// 1024 waves total to hide HBM latency on the 268 MB x stream (the ~12 us
// roofline floor at 23.3 TB/s).
// A (16x4 f32): lanes 0-15 hold rows M=0..15 at K={0,1}; lanes 16-31 same rows
//   at K={2,3} -> per-lane contiguous float2 at x + row*H + k + 2*(lane>>4).
// B (4x16 f32): mirror layout; B[k][n] = W[n][k] -> identical float2 pattern
//   on W rows (experts). W is 1 MB -> L0/L2 resident, refetch is cheap.
// Outer loop steps one 128B line per row; one prefetch per iteration runs
// ~8 lines (1 KB/row) ahead, offset clamped in-bounds via scalar min so a
// WGP-scope (non-speculative) prefetch can never touch an unmapped page.
// ---------------------------------------------------------------------------
__global__ void __launch_bounds__(128)
router_gemm_wmma(const float* __restrict__ x,
                 const float* __restrict__ W,
                 float* __restrict__ logits) {
    const int lane = threadIdx.x & 31;
    const int wave = threadIdx.x >> 5;
    const int row_base = blockIdx.x * 64 + wave * 16;    // 16 rows per wave

    const int r    = lane & 15;        // row/expert within 16-group
    const int koff = (lane >> 4) * 2;  // K sub-offset: 0 or 2

    const float* xp = x + (size_t)(row_base + r) * HDIM + koff;   // this wave's rows
    const float* wp = W + (size_t)r * HDIM + koff;                // expert r of each tile

    v8f acc[4] = {v8f{}, v8f{}, v8f{}, v8f{}};

    for (int k = 0; k < HDIM; k += 32) {
        // Stream prefetch, one 128B line per row, ~1 KB ahead of consumption.
        // Clamped so the address stays inside this row's x allocation.
        int kpf = k + 256;
        kpf = (kpf < HDIM - 32) ? kpf : (HDIM - 32);
        __builtin_prefetch(xp + kpf, 0, 3);
#pragma unroll
        for (int ks = 0; ks < 32; ks += 4) {
            const v2f a = ldg_nt2(xp + k + ks);
#pragma unroll
            for (int t = 0; t < 4; ++t) {
                const v2f b = ldg2(wp + (size_t)(t * 16) * HDIM + k + ks);
                acc[t] = __builtin_amdgcn_wmma_f32_16x16x4_f32(
                    false, a, false, b, (short)0, acc[t], false, false);
            }
        }
    }

    // C/D layout: lane -> N = lane&15, VGPR j -> M = j + 8*(lane>>4).
    const int col  = lane & 15;
    const int rgrp = lane >> 4;
#pragma unroll
    for (int t = 0; t < 4; ++t) {
#pragma unroll
        for (int j = 0; j < 8; ++j) {
            const int m  = rgrp * 8 + j;
            logits[(size_t)(row_base + m) * NEXP + t * 16 + col] = acc[t][j];
        }
    }
}

// ---------------------------------------------------------------------------
// Kernel 2: per-row softmax over 64 logits (in place), top-2 selection with
// lower-index tie-break, L1-normalized weights. One row per wave32.
// ---------------------------------------------------------------------------
static __device__ __forceinline__ bool better(float v, int vi, float w, int wi) {
    return (v > w) || (v == w && vi < wi);
}

__global__ void __launch_bounds__(256)
router_softmax_top2(float* __restrict__ scores,
                    float* __restrict__ weights,
                    int* __restrict__ experts) {
    const int lane = threadIdx.x & 31;
    const int wave = threadIdx.x >> 5;
    const int row  = blockIdx.x * 8 + wave;
    if (row >= NTOK) return;

    float* srow = scores + (size_t)row * NEXP;
    float v0 = srow[lane];
    float v1 = srow[lane + 32];

    // max over 64
    float m = fmaxf(v0, v1);
#pragma unroll
    for (int mask = 16; mask >= 1; mask >>= 1)
        m = fmaxf(m, __shfl_xor(m, mask, 32));

    float e0 = expf(v0 - m);
    float e1 = expf(v1 - m);
    float s = e0 + e1;
#pragma unroll
    for (int mask = 16; mask >= 1; mask >>= 1)
        s += __shfl_xor(s, mask, 32);
    const float inv = 1.0f / s;
    const float s0 = e0 * inv;
    const float s1 = e1 * inv;
    srow[lane]      = s0;
    srow[lane + 32] = s1;

    // per-lane top-2 (value, expert-id)
    float b1, b2; int i1, i2;
    if (better(s0, lane, s1, lane + 32)) { b1 = s0; i1 = lane;      b2 = s1; i2 = lane + 32; }
    else                                 { b1 = s1; i1 = lane + 32; b2 = s0; i2 = lane; }

    // butterfly merge across the wave
#pragma unroll
    for (int mask = 16; mask >= 1; mask >>= 1) {
        const float ob1 = __shfl_xor(b1, mask, 32);
        const int   oi1 = __shfl_xor(i1, mask, 32);
        const float ob2 = __shfl_xor(b2, mask, 32);
        const int   oi2 = __shfl_xor(i2, mask, 32);
        if (better(ob1, oi1, b1, i1)) {
            if (better(b1, i1, ob2, oi2)) { b2 = b1; i2 = i1; }
            else                          { b2 = ob2; i2 = oi2; }
            b1 = ob1; i1 = oi1;
        } else if (better(ob1, oi1, b2, i2)) {
            b2 = ob1; i2 = oi1;
        }
    }

    if (lane == 0) {
        const float norm = 1.0f / (b1 + b2);   // scores are positive -> L1 norm
        weights[(size_t)row * 2 + 0] = b1 * norm;
        weights[(size_t)row * 2 + 1] = b2 * norm;
        experts[(size_t)row * 2 + 0] = i1;
        experts[(size_t)row * 2 + 1] = i2;
    }
}

// ---------------------------------------------------------------------------
extern "C" void kernel_launch(void* const* d_in, const int* in_sizes, int n_in,
                              void* d_out, int out_size, void* d_ws, size_t ws_size,
                              hipStream_t stream) {
    const float* x = (const float*)d_in[0];
    const float* W = (const float*)d_in[1];
    // d_in[2] = top_k (always 2 for this reference)

    float* out     = (float*)d_out;
    float* scores  = out;                                   // [N, E]
    float* weights = out + (size_t)NTOK * NEXP;             // [N, 2]
    int*   experts = (int*)(out + (size_t)NTOK * NEXP + (size_t)NTOK * 2);  // [N, 2]

    // GEMM: 64 rows per block (4 waves x 16 rows), 256 blocks -> 1024 waves.
    router_gemm_wmma<<<NTOK / 64, 128, 0, stream>>>(x, W, scores);
    // Softmax + top-2: 8 rows per block (one row per wave).
    router_softmax_top2<<<NTOK / 8, 256, 0, stream>>>(scores, weights, experts);
}